// LorentzSparseSqDisAtt_4277787427327
// MI455X (gfx1250) — compile-verified
//
#include <hip/hip_runtime.h>
#include <hip/hip_bf16.h>
#include <math.h>

// ---------------------------------------------------------------------------
// Lorentz sparse squared-distance attention, CDNA5 (gfx1250, wave32).
//   Phase 1 (node): u = log0_tail(x); mu = u @ W^T + b; y = exp0(mu)
//                   GEMM done with V_WMMA_F32_16X16X4_F32 (full f32 precision;
//                   GEMM is bandwidth-bound so fp32 WMMA costs nothing).
//   Phase 2 (edge): Lorentz inner product over gathered rows of y (L2-resident).
// ---------------------------------------------------------------------------

typedef __attribute__((ext_vector_type(2))) float v2f;
typedef __attribute__((ext_vector_type(8))) float v8f;

#define DSP   64      // spatial dim
#define DP1   65      // x row length
#define YSTR  68      // y row stride (pad: head at [0], tail at [4..67], 16B aligned)

// One wave handles a 16-row tile: full 16x64 mu block via 64 WMMA f32 16x16x4.
__global__ __launch_bounds__(256) void lorentz_node_kernel(
    const float* __restrict__ x,     // [n_nodes][65]
    const float* __restrict__ W,     // [64][64] row-major (weight)
    const float* __restrict__ bias,  // [64]
    float* __restrict__ y,           // [n_nodes][YSTR] workspace
    int n_nodes)
{
    const int lane  = threadIdx.x & 31;
    const int wave  = threadIdx.x >> 5;          // 0..7
    const int half  = lane >> 4;                 // 0: K even pair, 1: K odd pair
    const int l16   = lane & 15;
    const int row0  = (blockIdx.x * 8 + wave) * 16;

    // --- load tail fragments in native f32 A-matrix layout ------------------
    // A 16x4 f32 layout: lanes 0-15 hold (K0,K1) of row M=lane, lanes 16-31 hold (K2,K3).
    const int node    = row0 + l16;
    const int nodeCl  = node < n_nodes ? node : (n_nodes - 1);   // clamp: keep EXEC full
    const float* xrow = x + (size_t)nodeCl * DP1;

    const float x0 = xrow[0];

    v2f a[16];
    float nrm2p = 0.0f;
#pragma unroll
    for (int kb = 0; kb < 16; ++kb) {
        // element offsets within tail: 4*kb + 2*half + {0,1}  (only DWORD aligned -> scalar loads)
        const float t0 = xrow[1 + 4 * kb + 2 * half];
        const float t1 = xrow[2 + 4 * kb + 2 * half];
        a[kb].x = t0;
        a[kb].y = t1;
        nrm2p += t0 * t0 + t1 * t1;
    }
    // combine the two half-rows (lane L <-> L^16 hold the same row m=l16)
    const float nrm2    = nrm2p + __shfl_xor(nrm2p, 16, 32);
    const float xt_norm = fmaxf(sqrtf(nrm2), 1e-10f);
    const float dist    = acoshf(fmaxf(x0, 1.0f + 1e-7f));      // sqrt(C)=1
    const float uscale  = dist / xt_norm;
#pragma unroll
    for (int kb = 0; kb < 16; ++kb) { a[kb].x *= uscale; a[kb].y *= uscale; }

    // --- GEMM: mu(16x64) = u(16x64) * W^T(64x64), via 16 K-steps x 4 N-tiles -
    v8f acc[4] = {};
#pragma unroll
    for (int kb = 0; kb < 16; ++kb) {
#pragma unroll
        for (int nb = 0; nb < 4; ++nb) {
            // B 4x16 (KxN): B[k][n] = W[n][k]; same half-wave K striping as A.
            // W row-major: W[n*64 + k], (k = 4*kb + 2*half) even -> 8B aligned float2.
            const float2 wb = *(const float2*)(W + (size_t)(nb * 16 + l16) * DSP
                                                 + 4 * kb + 2 * half);
            v2f b; b.x = wb.x; b.y = wb.y;
            acc[nb] = __builtin_amdgcn_wmma_f32_16x16x4_f32(
                /*neg_a=*/false, a[kb], /*neg_b=*/false, b,
                /*c_mod=*/(short)0, acc[nb], /*reuse_a=*/false, /*reuse_b=*/false);
        }
    }

    // --- bias + exp0 + store -------------------------------------------------
    float bv[4];
#pragma unroll
    for (int nb = 0; nb < 4; ++nb) bv[nb] = bias[nb * 16 + l16];

#pragma unroll
    for (int g = 0; g < 8; ++g) {
        // C/D layout: vgpr g, lanes 0-15 -> row g; lanes 16-31 -> row g+8; col = nb*16+l16
        float v0 = acc[0][g] + bv[0];
        float v1 = acc[1][g] + bv[1];
        float v2 = acc[2][g] + bv[2];
        float v3 = acc[3][g] + bv[3];
        float sq = v0 * v0 + v1 * v1 + v2 * v2 + v3 * v3;
        // reduce across the 16 lanes holding this row (xor<16 stays in-half)
        sq += __shfl_xor(sq, 1, 32);
        sq += __shfl_xor(sq, 2, 32);
        sq += __shfl_xor(sq, 4, 32);
        sq += __shfl_xor(sq, 8, 32);
        const float r     = fmaxf(sqrtf(sq), 1e-10f);
        const float head  = coshf(r);            // cosh(sqrtc*r)/sqrtc, sqrtc=1
        const float scale = sinhf(r) / r;        // sinh(sqrtc*r)/(sqrtc*r)

        const int outRow = row0 + g + 8 * half;
        if (outRow < n_nodes) {
            float* yr = y + (size_t)outRow * YSTR;
            yr[4 + 0 * 16 + l16] = v0 * scale;
            yr[4 + 1 * 16 + l16] = v1 * scale;
            yr[4 + 2 * 16 + l16] = v2 * scale;
            yr[4 + 3 * 16 + l16] = v3 * scale;
            if (l16 == 0) yr[0] = head;
        }
    }
}

// One thread per edge; y rows are L2-resident (13.6 MB << 192 MB L2).
__global__ __launch_bounds__(256) void lorentz_edge_kernel(
    const long long* __restrict__ adj,   // [2][n_edges] int64
    const float* __restrict__ y,         // [n_nodes][YSTR]
    float* __restrict__ out,             // [n_edges]
    int n_edges)
{
    const int e = blockIdx.x * blockDim.x + threadIdx.x;
    if (e >= n_edges) return;

    const long long s = adj[e];
    const long long d = adj[e + n_edges];
    const float* rs = y + (size_t)s * YSTR;
    const float* rd = y + (size_t)d * YSTR;

    float inner = -rs[0] * rd[0];
#pragma unroll
    for (int j = 0; j < 16; ++j) {
        const float4 ta = *(const float4*)(rs + 4 + 4 * j);   // 16B aligned (stride 68)
        const float4 tb = *(const float4*)(rd + 4 + 4 * j);
        inner += ta.x * tb.x + ta.y * tb.y + ta.z * tb.z + ta.w * tb.w;
    }
    float arg = -(1.0f + inner);                 // -(C + l_inner), C = 1
    arg = fminf(fmaxf(arg, 1e-10f), 1.0f);
    out[e] = expf(-arg);
}

extern "C" void kernel_launch(void* const* d_in, const int* in_sizes, int n_in,
                              void* d_out, int out_size, void* d_ws, size_t ws_size,
                              hipStream_t stream) {
    const float*     x    = (const float*)d_in[0];      // [n_nodes][65]
    const float*     W    = (const float*)d_in[1];      // [64][64]
    const float*     bias = (const float*)d_in[2];      // [64]
    const long long* adj  = (const long long*)d_in[3];  // [2][n_edges] int64
    float*           out  = (float*)d_out;
    float*           y    = (float*)d_ws;               // [n_nodes][68] scratch

    const int n_nodes = in_sizes[0] / DP1;
    const int n_edges = in_sizes[3] / 2;

    // Node phase: 8 waves/block * 16 rows/wave = 128 rows per block.
    const int nodeBlocks = (n_nodes + 127) / 128;
    lorentz_node_kernel<<<nodeBlocks, 256, 0, stream>>>(x, W, bias, y, n_nodes);

    // Edge phase.
    const int edgeBlocks = (n_edges + 255) / 256;
    lorentz_edge_kernel<<<edgeBlocks, 256, 0, stream>>>(adj, y, out, n_edges);
}